// CPABTransform_30769145709084
// MI455X (gfx1250) — compile-verified
//
#include <hip/hip_runtime.h>

typedef __attribute__((ext_vector_type(2))) float v2f;
typedef __attribute__((ext_vector_type(8))) float v8f;

#define NCELL 10
#define HDIM  10
#define DDIM  9
#define EPSC  1e-7f

__device__ __forceinline__ v8f wmma_k4(v2f a, v2f b, v8f c) {
  // V_WMMA_F32_16X16X4_F32 : D(16x16,f32) = A(16x4,f32) * B(4x16,f32) + C
  return __builtin_amdgcn_wmma_f32_16x16x4_f32(false, a, false, b, (short)0, c, false, false);
}

// ---- fold W3 and the CPAB basis projection into one 10x20 matrix ----
__global__ void cpab_precompute_kernel(const float* __restrict__ W3,
                                       const float* __restrict__ b3,
                                       const float* __restrict__ Bmat,
                                       float* __restrict__ ws) {
  int idx = threadIdx.x;
  if (idx < HDIM * 2 * NCELL) {                   // W3B[k][m] = sum_j W3[k,j]*B[m,j]
    int k = idx / (2 * NCELL), m = idx % (2 * NCELL);
    float s = 0.f;
    #pragma unroll
    for (int j = 0; j < DDIM; ++j) s += W3[k * DDIM + j] * Bmat[m * DDIM + j];
    ws[idx] = s;
  } else if (idx < HDIM * 2 * NCELL + 2 * NCELL) { // b3B[m] = sum_j b3[j]*B[m,j]
    int m = idx - HDIM * 2 * NCELL;
    float s = 0.f;
    #pragma unroll
    for (int j = 0; j < DDIM; ++j) s += b3[j] * Bmat[m * DDIM + j];
    ws[HDIM * 2 * NCELL + m] = s;
  }
}

__global__ __launch_bounds__(256, 2)
void cpab_forward_kernel(const float* __restrict__ x,
                         const float* __restrict__ W0, const float* __restrict__ b0,
                         const float* __restrict__ W1, const float* __restrict__ b1,
                         const float* __restrict__ W2, const float* __restrict__ b2,
                         const float* __restrict__ ws,
                         float* __restrict__ out, int N) {
  __shared__ float ldsT[8][16 * 17];   // per-wave 16x16 transpose tile (pad 17)
  __shared__ float ldsAB[8][32 * 21];  // per-wave 32 points x 20 coeffs (pad 21)

  const int lane = threadIdx.x & 31;
  const int w    = threadIdx.x >> 5;
  const int n    = lane & 15;   // N column / point-in-tile
  const int hi   = lane >> 4;   // half-wave select

  const long long base = ((long long)blockIdx.x * 8 + w) * 32;
  long long p  = base + lane;
  long long pc = (p < N) ? p : (long long)N - 1;

  // load + clip  (mask=[1,0]: x1 = x[:,1] -> MLP, x2 = x[:,0] -> integration)
  float x2v = fminf(fmaxf(x[pc * 2 + 0], EPSC), 1.f - EPSC);
  float x1v = fminf(fmaxf(x[pc * 2 + 1], EPSC), 1.f - EPSC);

  const float* W3B = ws;          // 10 x 20
  const float* b3B = ws + 200;    // 20

  // ---- weight fragments in B layout: lane n = column, VGPR j + hi pick K row ----
  v2f w1f[3], w2f[3], w3aF[3], w3bF[3];
  float w0v[3][2], b0v[3][2];
  #pragma unroll
  for (int c = 0; c < 3; ++c) {
    #pragma unroll
    for (int j = 0; j < 2; ++j) {
      int k = 4 * c + 2 * hi + j;
      bool kv = (k < HDIM);
      w1f[c][j]  = (kv && n < HDIM) ? W1[k * HDIM + n] : 0.f;
      w2f[c][j]  = (kv && n < HDIM) ? W2[k * HDIM + n] : 0.f;
      w3aF[c][j] = kv ? W3B[k * (2 * NCELL) + n] : 0.f;
      w3bF[c][j] = (kv && n < 4) ? W3B[k * (2 * NCELL) + 16 + n] : 0.f;
      w0v[c][j]  = kv ? W0[k] : 0.f;
      b0v[c][j]  = kv ? b0[k] : 0.f;
    }
  }
  float b1n  = (n < HDIM) ? b1[n] : 0.f;
  float b2n  = (n < HDIM) ? b2[n] : 0.f;
  float b3an = b3B[n];
  float b3bn = (n < 4) ? b3B[16 + n] : 0.f;

  // ---- MLP: two 16-point tiles per wave ----
  #pragma unroll
  for (int t = 0; t < 2; ++t) {
    float xm = __shfl(x1v, 16 * t + n, 32);   // x1 of this lane's A-row point

    // layer 0 directly in A layout: h0[m,k] = relu(x1*W0[k]+b0[k])
    v2f aF[3];
    #pragma unroll
    for (int c = 0; c < 3; ++c) {
      aF[c][0] = fmaxf(xm * w0v[c][0] + b0v[c][0], 0.f);
      aF[c][1] = fmaxf(xm * w0v[c][1] + b0v[c][1], 0.f);
    }

    // layer 1: 3 K-chunks of WMMA, then bias+relu in C/D layout
    v8f acc = {0.f, 0.f, 0.f, 0.f, 0.f, 0.f, 0.f, 0.f};
    #pragma unroll
    for (int c = 0; c < 3; ++c) acc = wmma_k4(aF[c], w1f[c], acc);
    #pragma unroll
    for (int r = 0; r < 8; ++r) acc[r] = fmaxf(acc[r] + b1n, 0.f);

    // transpose C/D -> A layout through LDS (same-wave, in-order DS)
    #pragma unroll
    for (int r = 0; r < 8; ++r) ldsT[w][(8 * hi + r) * 17 + n] = acc[r];
    asm volatile("s_wait_dscnt 0" ::: "memory");
    #pragma unroll
    for (int c = 0; c < 3; ++c) {
      aF[c][0] = ldsT[w][n * 17 + (4 * c + 2 * hi + 0)];
      aF[c][1] = ldsT[w][n * 17 + (4 * c + 2 * hi + 1)];
    }
    asm volatile("s_wait_dscnt 0" ::: "memory");

    // layer 2
    v8f acc2 = {0.f, 0.f, 0.f, 0.f, 0.f, 0.f, 0.f, 0.f};
    #pragma unroll
    for (int c = 0; c < 3; ++c) acc2 = wmma_k4(aF[c], w2f[c], acc2);
    #pragma unroll
    for (int r = 0; r < 8; ++r) acc2[r] = fmaxf(acc2[r] + b2n, 0.f);

    #pragma unroll
    for (int r = 0; r < 8; ++r) ldsT[w][(8 * hi + r) * 17 + n] = acc2[r];
    asm volatile("s_wait_dscnt 0" ::: "memory");
    #pragma unroll
    for (int c = 0; c < 3; ++c) {
      aF[c][0] = ldsT[w][n * 17 + (4 * c + 2 * hi + 0)];
      aF[c][1] = ldsT[w][n * 17 + (4 * c + 2 * hi + 1)];
    }

    // fused final layer: AB = h2 @ W3B + b3B  (20 cols -> two WMMAs)
    v8f dA = {0.f, 0.f, 0.f, 0.f, 0.f, 0.f, 0.f, 0.f};
    v8f dB = {0.f, 0.f, 0.f, 0.f, 0.f, 0.f, 0.f, 0.f};
    #pragma unroll
    for (int c = 0; c < 3; ++c) { dA = wmma_k4(aF[c], w3aF[c], dA);
                                  dB = wmma_k4(aF[c], w3bF[c], dB); }
    #pragma unroll
    for (int r = 0; r < 8; ++r) {
      int prow = 16 * t + 8 * hi + r;
      ldsAB[w][prow * 21 + n] = dA[r] + b3an;
      if (n < 4) ldsAB[w][prow * 21 + 16 + n] = dB[r] + b3bn;
    }
  }
  asm volatile("s_wait_dscnt 0" ::: "memory");

  // ---- CPAB integration: one point per lane, 11 fixed steps ----
  const float* AB = &ldsAB[w][lane * 21];   // a[c]=AB[2c], b[c]=AB[2c+1]
  float phi = x2v;
  int   c   = (int)floorf(phi * (float)NCELL);
  c = min(max(c, 0), NCELL - 1);
  float tt = 1.f, s = 0.f;
  bool  done = false;

  #pragma unroll 1
  for (int it = 0; it < NCELL + 1; ++it) {
    float a = AB[2 * c], b = AB[2 * c + 1];
    float v = a * phi + b;
    bool  big = fabsf(a) > 1e-10f;
    float a_safe = big ? a : 1.f;
    float boa = b / a_safe;
    float psi = big ? (phi + boa) * __expf(a * tt) - boa : phi + b * tt;
    float lo  = (float)c / (float)NCELL;
    float hif = (float)(c + 1) / (float)NCELL;
    bool  inside = (psi >= lo) && (psi <= hif);
    float xc = (v >= 0.f) ? hif : lo;
    float vb = a * xc + b;
    float v_safe = (v == 0.f) ? 1.f : v;
    float ratio = fmaxf(vb / v_safe, 1e-30f);
    float t_hit = big ? __logf(ratio) / a_safe : (xc - phi) / v_safe;
    bool  act   = !done;
    bool  cross = act && !inside;
    s  += act ? a * (inside ? tt : t_hit) : 0.f;
    phi = cross ? xc : ((act && inside) ? psi : phi);
    tt  = cross ? tt - t_hit : tt;
    int cn = cross ? c + ((v >= 0.f) ? 1 : -1) : c;
    c = min(max(cn, 0), NCELL - 1);
    done = done || inside;
  }

  // ---- outputs: z[:,0]=z2, z[:,1]=x1 ; logdet[:,0]=s, [:,1]=0 ----
  if (p < N) {
    float2* z  = (float2*)out;
    float2* ld = (float2*)(out + 2 * (long long)N);
    z[p]  = make_float2(phi, x1v);
    ld[p] = make_float2(s, 0.f);
  }
}

extern "C" void kernel_launch(void* const* d_in, const int* in_sizes, int n_in,
                              void* d_out, int out_size, void* d_ws, size_t ws_size,
                              hipStream_t stream) {
  const float* x  = (const float*)d_in[0];
  const float* B  = (const float*)d_in[1];
  const float* W0 = (const float*)d_in[2];
  const float* b0 = (const float*)d_in[3];
  const float* W1 = (const float*)d_in[4];
  const float* b1 = (const float*)d_in[5];
  const float* W2 = (const float*)d_in[6];
  const float* b2 = (const float*)d_in[7];
  const float* W3 = (const float*)d_in[8];
  const float* b3 = (const float*)d_in[9];
  float* out = (float*)d_out;
  float* ws  = (float*)d_ws;
  int N = in_sizes[0] / 2;

  cpab_precompute_kernel<<<1, 256, 0, stream>>>(W3, b3, B, ws);

  long long waves  = ((long long)N + 31) / 32;
  int       blocks = (int)((waves + 7) / 8);
  cpab_forward_kernel<<<blocks, 256, 0, stream>>>(x, W0, b0, W1, b1, W2, b2, ws, out, N);
}